// MSPSurfNet_61495341744443
// MI455X (gfx1250) — compile-verified
//
#include <hip/hip_runtime.h>

typedef __attribute__((ext_vector_type(2))) float v2f;
typedef __attribute__((ext_vector_type(8))) float v8f;

#define MQ 2048
#define NN 32768
#define DCH 256

// ---------------------------------------------------------------------------
// Kernel 1: nearest-neighbor argmin via V_WMMA_F32_16X16X4_F32.
// One wave handles one (search, 16-query tile). For each 16-node tile:
//   A[16x4] = query coords (x,y,z,0), B[4x16] = node coords (x,y,z,0)
//   D = A x B  -> 256 dot products; score = |p|^2 - 2*dot (|q|^2 constant
//   per query, irrelevant to argmin). Running (min,idx) kept per D-row VGPR,
//   reduced across each 16-lane half at the end.
// ---------------------------------------------------------------------------
__global__ __launch_bounds__(256) void nn_argmin_kernel(
    const float* __restrict__ coords_orig,
    const float* __restrict__ coords_mut,
    const float* __restrict__ pos0, const float* __restrict__ pos1,
    const float* __restrict__ pos2, const float* __restrict__ pos3,
    int* __restrict__ idx_out)
{
    const int wave = blockIdx.x * (blockDim.x >> 5) + (threadIdx.x >> 5);
    const int srch = wave >> 7;       // 0..3
    const int tile = wave & 127;      // 0..127 query tiles of 16
    const int lane = threadIdx.x & 31;
    const int h    = lane >> 4;       // K-half (0: K=0,1  1: K=2,3)
    const int col  = lane & 15;       // M for A-load, N for B/D

    const float* Q = (srch < 2) ? coords_orig : coords_mut;
    const float* P = (srch == 0) ? pos0 : (srch == 1) ? pos1
                   : (srch == 2) ? pos2 : pos3;

    // A operand: lane (h,col) holds Q[col][2h], Q[col][2h+1]; pad K=3 with 0
    const int qbase = tile * 16;
    const int m = qbase + col;
    const float qx = Q[m * 3 + 0], qy = Q[m * 3 + 1], qz = Q[m * 3 + 2];
    v2f a;
    a.x = h ? qz : qx;
    a.y = h ? 0.0f : qy;

    float bestS[8];
    int   bestI[8];
#pragma unroll
    for (int r = 0; r < 8; ++r) { bestS[r] = 3.0e38f; bestI[r] = 0; }

#pragma unroll 2
    for (int nbase = 0; nbase < NN; nbase += 16) {
        const int n = nbase + col;
        const float px = P[n * 3 + 0], py = P[n * 3 + 1], pz = P[n * 3 + 2];
        // B operand: lane (h,col) holds P[col][2h], P[col][2h+1]
        v2f b;
        b.x = h ? pz : px;
        b.y = h ? 0.0f : py;
        const float pn2 = px * px + py * py + pz * pz;  // per-lane |p_n|^2

        v8f c = {};
        c = __builtin_amdgcn_wmma_f32_16x16x4_f32(
                /*neg_a=*/false, a, /*neg_b=*/false, b,
                /*c_mod=*/(short)0, c, /*reuse_a=*/false, /*reuse_b=*/false);

#pragma unroll
        for (int r = 0; r < 8; ++r) {
            // score for (M = r + 8*h in this wave-tile, N = nbase + col)
            const float s = __builtin_fmaf(-2.0f, c[r], pn2);
            const bool lt = s < bestS[r];
            bestI[r] = lt ? n : bestI[r];
            bestS[r] = lt ? s : bestS[r];
        }
    }

    // Reduce (min, idx) across the 16 lanes of each wave half.
    // Lane 0 ends with result for m = qbase+r, lane 16 for m = qbase+r+8.
#pragma unroll
    for (int r = 0; r < 8; ++r) {
        float bs = bestS[r];
        int   bi = bestI[r];
#pragma unroll
        for (int off = 1; off < 16; off <<= 1) {
            const float os = __shfl_xor(bs, off, 32);
            const int   oi = __shfl_xor(bi, off, 32);
            if (os < bs || (os == bs && oi < bi)) { bs = os; bi = oi; }
        }
        if (col == 0)
            idx_out[srch * MQ + qbase + r + 8 * h] = bi;
    }
}

// ---------------------------------------------------------------------------
// Kernel 2: gather feat rows by argmin index, average left/right, max over M.
// block 0 -> orig (feat0/feat1, idx0/idx1), block 1 -> mut (feat2/feat3).
// Thread t owns channel t; loads are coalesced 1KB rows.
// ---------------------------------------------------------------------------
__global__ __launch_bounds__(256) void gather_max_kernel(
    const float* __restrict__ feat0, const float* __restrict__ feat1,
    const float* __restrict__ feat2, const float* __restrict__ feat3,
    const int* __restrict__ idxbuf, float* __restrict__ emb)
{
    const int side = blockIdx.x;                  // 0 = orig, 1 = mut
    const float* fa = side ? feat2 : feat0;
    const float* fb = side ? feat3 : feat1;
    const int* ia = idxbuf + (side ? 2 : 0) * MQ;
    const int* ib = idxbuf + (side ? 3 : 1) * MQ;
    const int d = threadIdx.x;

    float acc = -3.0e38f;
    for (int m = 0; m < MQ; ++m) {
        const int i0 = ia[m];
        const int i1 = ib[m];
        const float v = 0.5f * (fa[i0 * DCH + d] + fb[i1 * DCH + d]);
        acc = fmaxf(acc, v);
    }
    emb[side * DCH + d] = acc;
}

// ---------------------------------------------------------------------------
// Kernel 3: x[512] -> FC1(512x512) + ReLU -> FC2(1x512) + bias -> scalar.
// ---------------------------------------------------------------------------
__global__ __launch_bounds__(512) void mlp_kernel(
    const float* __restrict__ emb,
    const float* __restrict__ fc1_w, const float* __restrict__ fc1_b,
    const float* __restrict__ fc2_w, const float* __restrict__ fc2_b,
    float* __restrict__ out)
{
    __shared__ float xs[512];
    __shared__ float red[512];
    const int j = threadIdx.x;
    xs[j] = emb[j];
    __syncthreads();

    float acc = fc1_b[j];
    for (int k = 0; k < 512; ++k)
        acc = __builtin_fmaf(xs[k], fc1_w[j * 512 + k], acc);
    const float hj = fmaxf(acc, 0.0f);

    red[j] = hj * fc2_w[j];
    __syncthreads();
    for (int s = 256; s > 0; s >>= 1) {
        if (j < s) red[j] += red[j + s];
        __syncthreads();
    }
    if (j == 0) out[0] = red[0] + fc2_b[0];
}

// ---------------------------------------------------------------------------
extern "C" void kernel_launch(void* const* d_in, const int* in_sizes, int n_in,
                              void* d_out, int out_size, void* d_ws, size_t ws_size,
                              hipStream_t stream)
{
    const float* coords_orig = (const float*)d_in[0];
    const float* coords_mut  = (const float*)d_in[1];
    const float* pos0  = (const float*)d_in[2];
    const float* pos1  = (const float*)d_in[3];
    const float* pos2  = (const float*)d_in[4];
    const float* pos3  = (const float*)d_in[5];
    const float* feat0 = (const float*)d_in[6];
    const float* feat1 = (const float*)d_in[7];
    const float* feat2 = (const float*)d_in[8];
    const float* feat3 = (const float*)d_in[9];
    const float* fc1_w = (const float*)d_in[10];
    const float* fc1_b = (const float*)d_in[11];
    const float* fc2_w = (const float*)d_in[12];
    const float* fc2_b = (const float*)d_in[13];
    float* out = (float*)d_out;

    int*   idxbuf = (int*)d_ws;                                   // 4*2048 ints
    float* emb    = (float*)((char*)d_ws + 4 * MQ * sizeof(int)); // 512 floats

    // 512 wave-jobs = 4 searches x 128 query tiles; 8 waves per 256-thread block
    nn_argmin_kernel<<<64, 256, 0, stream>>>(coords_orig, coords_mut,
                                             pos0, pos1, pos2, pos3, idxbuf);
    gather_max_kernel<<<2, 256, 0, stream>>>(feat0, feat1, feat2, feat3,
                                             idxbuf, emb);
    mlp_kernel<<<1, 512, 0, stream>>>(emb, fc1_w, fc1_b, fc2_w, fc2_b, out);
}